// SelfAttention_61684320305351
// MI455X (gfx1250) — compile-verified
//
#include <hip/hip_runtime.h>
#include <math.h>

// ---------------- problem constants ----------------
#define D_MODEL 4096
#define HQ      32
#define HKV     8
#define NREP    4
#define HD      128
#define BATCH   8
#define KV_PAST 4095
#define T_TOT   4096
#define SPLITS  8
#define CHUNK_T (T_TOT / SPLITS)      // 512 tokens per partial block
#define N_QKV   (HQ*HD + 2*HKV*HD)    // 6144 fused QKV output columns
#define KC      512                   // K-chunk staged in LDS
#define XSTR    (KC + 8)              // padded LDS row stride (floats), 520%64=8 -> no bank conflicts

typedef __attribute__((ext_vector_type(2))) float v2f;
typedef __attribute__((ext_vector_type(8))) float v8f;

// ---------------------------------------------------------------------------
// M=8 (padded to 16) x N=16 tile GEMM over K=4096 using V_WMMA_F32_16X16X4_F32.
// A fragment (16x4 f32): lanes 0-15 hold K=k,k+1 ; lanes 16-31 hold K=k+2,k+3.
// B fragment (4x16 f32): lanes 0-15 hold rows K=k,k+1 for N=lane ; lanes 16-31 rows k+2,k+3.
// LDS tile is 16 rows: rows 8..15 are zeroed ONCE so every lane loads
// unconditionally (no EXEC save/restore in the hot loop). 8 A/B fragment
// pairs are fetched into registers before the 8 dependent WMMAs so 16 loads
// are in flight per batch instead of 1.
// Assumes blockDim.x == 128 (4 waves).
// ---------------------------------------------------------------------------
__device__ __forceinline__ v8f gemm_m8n16(const float* __restrict__ X,
                                          const float* __restrict__ wrow,
                                          float* __restrict__ xs,
                                          const int tid, const int lane)
{
    const int koff = (lane >= 16) ? 2 : 0;
    const int arow = lane & 15;

    // one-time zero fill of padded rows 8..15 (never rewritten afterwards)
    #pragma unroll
    for (int i = 0; i < 8; ++i) {
        const int vi   = i * 128 + tid;       // 0..1023 float4 slots
        const int row  = 8 + (vi >> 7);
        const int colv = vi & 127;
        *(float4*)(&xs[row * XSTR + colv * 4]) = make_float4(0.f, 0.f, 0.f, 0.f);
    }

    v8f c = {};
    for (int kc = 0; kc < D_MODEL; kc += KC) {
        __syncthreads();
        // cooperative stage of x[0..7][kc..kc+KC) into LDS, 8 float4 per thread
        #pragma unroll
        for (int i = 0; i < 8; ++i) {
            const int vi   = i * 128 + tid;
            const int row  = vi >> 7;           // 0..7
            const int colv = vi & 127;
            const float4 val = *(const float4*)(X + (size_t)row * D_MODEL + kc + colv * 4);
            *(float4*)(&xs[row * XSTR + colv * 4]) = val;
        }
        __syncthreads();

        const float* wp = wrow + kc;
        if (kc + KC < D_MODEL)
            __builtin_prefetch(wp + KC, 0, 0);   // global_prefetch_b8 next chunk

        for (int kk = 0; kk < KC; kk += 32) {
            v2f a[8], b[8];
            #pragma unroll
            for (int u = 0; u < 8; ++u) {
                a[u] = *(const v2f*)(&xs[arow * XSTR + kk + u * 4 + koff]); // ds_load_b64
                b[u] = *(const v2f*)(wp + kk + u * 4 + koff);               // global_load_b64
            }
            #pragma unroll
            for (int u = 0; u < 8; ++u)
                c = __builtin_amdgcn_wmma_f32_16x16x4_f32(
                        /*neg_a=*/false, a[u], /*neg_b=*/false, b[u],
                        /*c_mod=*/(short)0, c, /*reuse_a=*/false, /*reuse_b=*/false);
        }
    }
    return c;
}

// ---------------------------------------------------------------------------
// Kernel 1: fused QKV projection + RoPE.
// One wave owns one 16-wide column tile of the 6144 fused outputs.
// grid = 96 blocks x 128 threads (4 waves) -> 384 tiles.
// C layout: VGPR i, lanes 0-15 hold out[batch=i][n0+lane] (M=0..7 valid).
// RoPE pairs (even,odd dims) live in adjacent lanes -> shfl_xor(1).
// ---------------------------------------------------------------------------
__global__ void __launch_bounds__(128)
qkv_rope_kernel(const float* __restrict__ x,
                const float* __restrict__ fcos, const float* __restrict__ fsin,
                const float* __restrict__ wq, const float* __restrict__ wk,
                const float* __restrict__ wv,
                float* __restrict__ qbuf, float* __restrict__ knew,
                float* __restrict__ vnew)
{
    __shared__ float xs[16 * XSTR];
    const int tid  = threadIdx.x;
    const int wave = tid >> 5;
    const int lane = tid & 31;
    const int ct   = blockIdx.x * 4 + wave;   // 0..383
    const int n0   = ct * 16;

    const float* W;
    int rowbase;
    if      (n0 < HQ*HD)          { W = wq; rowbase = n0; }
    else if (n0 < HQ*HD + HKV*HD) { W = wk; rowbase = n0 - HQ*HD; }
    else                          { W = wv; rowbase = n0 - (HQ*HD + HKV*HD); }
    const float* wrow = W + (size_t)(rowbase + (lane & 15)) * D_MODEL;

    v8f c = gemm_m8n16(x, wrow, xs, tid, lane);

    // ---- RoPE + writeback (lanes 0-15 hold valid data) ----
    const int   ng  = n0 + (lane & 15);
    const int   dd  = ng & (HD - 1);
    const float cj  = fcos[dd >> 1];
    const float sj  = fsin[dd >> 1];
    const bool  isv = (n0 >= HQ*HD + HKV*HD);
    const bool  odd = (dd & 1);

    #pragma unroll
    for (int i = 0; i < BATCH; ++i) {
        const float val     = c[i];
        const float partner = __shfl_xor(val, 1, 32);   // paired rotary dim
        float res;
        if (isv) res = val;
        else     res = odd ? (partner * sj + val * cj) : (val * cj - partner * sj);
        if (lane < 16) {
            if (n0 < HQ*HD)  qbuf[(size_t)i * (HQ*HD)  + ng] = res;
            else if (!isv)   knew[(size_t)i * (HKV*HD) + (ng - HQ*HD)] = res;
            else             vnew[(size_t)i * (HKV*HD) + (ng - HQ*HD - HKV*HD)] = res;
        }
    }
}

// ---------------------------------------------------------------------------
// Kernel 2: flash-decoding partials. block = (b, kv-head g, T-chunk).
// Each wave owns 64 tokens: lanes stripe HD (4 floats/lane -> coalesced b128
// K/V row loads), per-token dot for the 4 grouped Q heads + wave reduction,
// online softmax with shared running max, then cross-wave merge in LDS.
// ---------------------------------------------------------------------------
__device__ __forceinline__ float wred(float v) {
    #pragma unroll
    for (int o = 16; o > 0; o >>= 1) v += __shfl_xor(v, o, 32);
    return v;
}

__global__ void __launch_bounds__(256)
attn_partial_kernel(const float* __restrict__ qbuf,
                    const float* __restrict__ knew, const float* __restrict__ vnew,
                    const float* __restrict__ k_past, const float* __restrict__ v_past,
                    float* __restrict__ pm, float* __restrict__ pl,
                    float* __restrict__ pacc)
{
    const int blk   = blockIdx.x;
    const int chunk = blk & (SPLITS - 1);
    const int g     = (blk / SPLITS) & (HKV - 1);
    const int b     = blk / (SPLITS * HKV);
    const int tid   = threadIdx.x;
    const int wave  = tid >> 5;
    const int lane  = tid & 31;

    float qr[NREP][4];
    #pragma unroll
    for (int r = 0; r < NREP; ++r) {
        const float4 q4 = *(const float4*)(qbuf + ((size_t)b*HQ + g*NREP + r) * HD + lane*4);
        qr[r][0] = q4.x; qr[r][1] = q4.y; qr[r][2] = q4.z; qr[r][3] = q4.w;
    }

    const float scale = 0.08838834764831845f;   // 1/sqrt(128)
    const size_t kv_bg = (size_t)b * HKV + g;
    const float* kp = k_past + kv_bg * (size_t)KV_PAST * HD;
    const float* vp = v_past + kv_bg * (size_t)KV_PAST * HD;

    float m = -1e30f;
    float l[NREP] = {0.f, 0.f, 0.f, 0.f};
    float acc[NREP][4] = {};

    for (int t = chunk * CHUNK_T + wave; t < (chunk + 1) * CHUNK_T; t += 8) {
        const float4 k4 = (t < KV_PAST)
            ? *(const float4*)(kp + (size_t)t * HD + lane*4)
            : *(const float4*)(knew + kv_bg * HD + lane*4);

        float s[NREP];
        #pragma unroll
        for (int r = 0; r < NREP; ++r) {
            float d = qr[r][0]*k4.x + qr[r][1]*k4.y + qr[r][2]*k4.z + qr[r][3]*k4.w;
            s[r] = wred(d) * scale;
        }
        const float smax = fmaxf(fmaxf(s[0], s[1]), fmaxf(s[2], s[3]));
        const float mnew = fmaxf(m, smax);
        const float rf   = __expf(m - mnew);

        const float4 v4 = (t < KV_PAST)
            ? *(const float4*)(vp + (size_t)t * HD + lane*4)
            : *(const float4*)(vnew + kv_bg * HD + lane*4);

        #pragma unroll
        for (int r = 0; r < NREP; ++r) {
            const float p = __expf(s[r] - mnew);
            l[r] = l[r] * rf + p;
            acc[r][0] = acc[r][0]*rf + p*v4.x;
            acc[r][1] = acc[r][1]*rf + p*v4.y;
            acc[r][2] = acc[r][2]*rf + p*v4.z;
            acc[r][3] = acc[r][3]*rf + p*v4.w;
        }
        m = mnew;
    }

    // ---- merge 8 waves ----
    __shared__ float sm[8];
    __shared__ float sl[8][NREP];
    __shared__ float sacc[8][NREP * HD];   // 16 KB
    if (lane == 0) {
        sm[wave] = m;
        #pragma unroll
        for (int r = 0; r < NREP; ++r) sl[wave][r] = l[r];
    }
    #pragma unroll
    for (int r = 0; r < NREP; ++r)
        #pragma unroll
        for (int j = 0; j < 4; ++j)
            sacc[wave][r * HD + lane * 4 + j] = acc[r][j];
    __syncthreads();

    float mg = sm[0];
    #pragma unroll
    for (int w = 1; w < 8; ++w) mg = fmaxf(mg, sm[w]);

    for (int slot = tid; slot < NREP * HD; slot += 256) {
        float a = 0.f;
        #pragma unroll
        for (int w = 0; w < 8; ++w) a += sacc[w][slot] * __expf(sm[w] - mg);
        pacc[(size_t)blk * NREP * HD + slot] = a;
    }
    if (tid < NREP) {
        float lr = 0.f;
        #pragma unroll
        for (int w = 0; w < 8; ++w) lr += sl[w][tid] * __expf(sm[w] - mg);
        pl[blk * NREP + tid] = lr;
    }
    if (tid == 0) pm[blk] = mg;
}

// ---------------------------------------------------------------------------
// Kernel 3: combine the SPLITS partials per (b, q-head) -> attn_out.
// ---------------------------------------------------------------------------
__global__ void __launch_bounds__(HD)
attn_reduce_kernel(const float* __restrict__ pm, const float* __restrict__ pl,
                   const float* __restrict__ pacc, float* __restrict__ attn)
{
    const int blk = blockIdx.x;          // b*HQ + h
    const int h   = blk & (HQ - 1);
    const int b   = blk / HQ;
    const int g   = h / NREP;
    const int r   = h & (NREP - 1);
    const int d   = threadIdx.x;
    const int base = (b * HKV + g) * SPLITS;

    float mg = -1e30f;
    #pragma unroll
    for (int c = 0; c < SPLITS; ++c) mg = fmaxf(mg, pm[base + c]);

    float lg = 0.f, a = 0.f;
    #pragma unroll
    for (int c = 0; c < SPLITS; ++c) {
        const float w = __expf(pm[base + c] - mg);
        lg += pl[(base + c) * NREP + r] * w;
        a  += pacc[(size_t)(base + c) * NREP * HD + r * HD + d] * w;
    }
    attn[((size_t)b * HQ + h) * HD + d] = a / lg;
}

// ---------------------------------------------------------------------------
// Kernel 4: output projection  out = attn(8x4096) @ wo^T  via WMMA f32.
// grid = 64 blocks x 128 threads -> 256 column tiles.
// ---------------------------------------------------------------------------
__global__ void __launch_bounds__(128)
outproj_kernel(const float* __restrict__ attn, const float* __restrict__ wo,
               float* __restrict__ out)
{
    __shared__ float xs[16 * XSTR];
    const int tid  = threadIdx.x;
    const int wave = tid >> 5;
    const int lane = tid & 31;
    const int ct   = blockIdx.x * 4 + wave;   // 0..255
    const int n0   = ct * 16;

    const float* wrow = wo + (size_t)(n0 + (lane & 15)) * D_MODEL;
    v8f c = gemm_m8n16(attn, wrow, xs, tid, lane);

    const int ng = n0 + (lane & 15);
    if (lane < 16) {
        #pragma unroll
        for (int i = 0; i < BATCH; ++i)
            out[(size_t)i * D_MODEL + ng] = c[i];
    }
}

// ---------------------------------------------------------------------------
extern "C" void kernel_launch(void* const* d_in, const int* in_sizes, int n_in,
                              void* d_out, int out_size, void* d_ws, size_t ws_size,
                              hipStream_t stream)
{
    const float* x      = (const float*)d_in[0];
    const float* fcos   = (const float*)d_in[1];
    const float* fsin   = (const float*)d_in[2];
    const float* k_past = (const float*)d_in[3];
    const float* v_past = (const float*)d_in[4];
    const float* wq     = (const float*)d_in[5];
    const float* wk     = (const float*)d_in[6];
    const float* wv     = (const float*)d_in[7];
    const float* wo     = (const float*)d_in[8];
    float* out = (float*)d_out;

    float* ws   = (float*)d_ws;
    float* qbuf = ws;              // 8*32*128          = 32768
    float* knew = qbuf + 32768;    // 8*8*128           = 8192
    float* vnew = knew + 8192;     // 8*8*128           = 8192
    float* attn = vnew + 8192;     // 8*32*128          = 32768
    float* pm   = attn + 32768;    // 8*8*8             = 512
    float* pl   = pm + 512;        // 512*4             = 2048
    float* pacc = pl + 2048;       // 512*4*128         = 262144  (~1.36 MB total)

    qkv_rope_kernel<<<N_QKV / 16 / 4, 128, 0, stream>>>(x, fcos, fsin, wq, wk, wv,
                                                        qbuf, knew, vnew);
    attn_partial_kernel<<<BATCH * HKV * SPLITS, 256, 0, stream>>>(qbuf, knew, vnew,
                                                                  k_past, v_past,
                                                                  pm, pl, pacc);
    attn_reduce_kernel<<<BATCH * HQ, HD, 0, stream>>>(pm, pl, pacc, attn);
    outproj_kernel<<<D_MODEL / 16 / 4, 128, 0, stream>>>(attn, wo, out);
}